// GraphSAGE_79577154060604
// MI455X (gfx1250) — compile-verified
//
#include <hip/hip_runtime.h>

#define NNODES 4096
#define HDIM   128
#define THR    0.99f
#define EPS    1e-12f

typedef __attribute__((ext_vector_type(2))) float v2f;
typedef __attribute__((ext_vector_type(8))) float v8f;

// ---------------------------------------------------------------------------
// Kernel 1: compress A (fp32, 64 MB) into a bit mask (2 MB). One wave makes
// one 32-bit word per iteration via ballot; fully coalesced 128B/wave loads.
// ---------------------------------------------------------------------------
__global__ void build_mask_kernel(const float* __restrict__ A,
                                  unsigned* __restrict__ maskw)
{
    const int lane  = threadIdx.x & 31;
    const int gwave = (blockIdx.x * blockDim.x + threadIdx.x) >> 5;
    const int nwaves = (gridDim.x * blockDim.x) >> 5;
    const int totalWords = NNODES * (NNODES / 32);

    for (int w = gwave; w < totalWords; w += nwaves) {
        const int i  = w >> 7;          // row (NNODES/32 = 128 words per row)
        const int wc = w & 127;         // word within row
        float a = A[(size_t)i * NNODES + wc * 32 + lane];
        unsigned long long b = __ballot(a > THR);   // wave32: low 32 bits
        if (lane == 0) maskw[w] = (unsigned)b;
    }
}

// ---------------------------------------------------------------------------
// Kernel 2: out[M x 128] = act(X[M x K] @ W[K x 128] + bias), act = relu?
// One wave computes one 16x16 tile with V_WMMA_F32_16X16X4_F32 (exact fp32).
// Fragment layout per CDNA5 ISA:
//   A 16x4:  lanes 0-15 -> M=lane, {v0,v1} = K=0,1 ; lanes 16-31 -> K=2,3
//   B 4x16:  lanes 0-15 -> N=lane, {v0,v1} = K=0,1 ; lanes 16-31 -> K=2,3
//   C/D:     VGPR r -> row r (lanes 0-15) / row r+8 (lanes 16-31), col=lane%16
// ---------------------------------------------------------------------------
__global__ void gemm_bias_act_kernel(const float* __restrict__ X,
                                     const float* __restrict__ W,
                                     const float* __restrict__ bias,
                                     float* __restrict__ out,
                                     int M, int K, int do_relu)
{
    const int lane = threadIdx.x & 31;
    const int wave = (blockIdx.x * blockDim.x + threadIdx.x) >> 5;
    const int tilesN = HDIM / 16;                 // 8 column tiles
    const int tileM  = wave / tilesN;
    const int tileN  = wave % tilesN;
    if (tileM >= (M >> 4)) return;                // wave-uniform; grid is exact

    const int row0 = tileM * 16;
    const int col0 = tileN * 16;
    const int ml   = lane & 15;
    const int kh   = (lane >> 4) << 1;            // 0 or 2

    v8f acc = {};
    const float* aBase = X + (size_t)(row0 + ml) * K + kh;
    const float* bBase = W + (size_t)kh * HDIM + col0 + ml;

    for (int k0 = 0; k0 < K; k0 += 4) {
        v2f a; a.x = aBase[k0];                 a.y = aBase[k0 + 1];
        v2f b; b.x = bBase[(size_t)k0 * HDIM];  b.y = bBase[(size_t)(k0 + 1) * HDIM];
        acc = __builtin_amdgcn_wmma_f32_16x16x4_f32(
            /*neg_a=*/false, a, /*neg_b=*/false, b,
            /*c_mod=*/(short)0, acc, /*reuse_a=*/false, /*reuse_b=*/false);
    }

    const int c     = col0 + ml;
    const int rbase = row0 + ((lane >> 4) << 3);  // +0 or +8
    const float bv  = bias[c];
#pragma unroll
    for (int r = 0; r < 8; ++r) {
        float v = acc[r] + bv;
        if (do_relu) v = fmaxf(v, 0.0f);
        out[(size_t)(rbase + r) * HDIM + c] = v;
    }
}

// ---------------------------------------------------------------------------
// Kernel 3: masked neighbor max-pool. One wave per destination node; wave
// scans 128 mask words; for each neighbor bit the 32 lanes float4-load the
// 128-wide score row and fmax into registers. scores >= 0 (relu), so acc=0
// init exactly matches the "no neighbors -> 0" reference semantics.
// ---------------------------------------------------------------------------
__global__ void maxpool_kernel(const unsigned* __restrict__ maskw,
                               const float* __restrict__ scores,
                               float* __restrict__ pooled)
{
    const int lane = threadIdx.x & 31;
    const int node = (blockIdx.x * blockDim.x + threadIdx.x) >> 5;
    if (node >= NNODES) return;

    float a0 = 0.f, a1 = 0.f, a2 = 0.f, a3 = 0.f;
    const unsigned* mrow = maskw + (size_t)node * (NNODES / 32);

    for (int w = 0; w < NNODES / 32; ++w) {
        unsigned bits = mrow[w];
        while (bits) {
            const int b = __builtin_ctz(bits);
            bits &= bits - 1;
            const int j = (w << 5) + b;
            const float4 s = *(const float4*)(scores + (size_t)j * HDIM + lane * 4);
            a0 = fmaxf(a0, s.x);
            a1 = fmaxf(a1, s.y);
            a2 = fmaxf(a2, s.z);
            a3 = fmaxf(a3, s.w);
        }
    }
    float4 o; o.x = a0; o.y = a1; o.z = a2; o.w = a3;
    *(float4*)(pooled + (size_t)node * HDIM + lane * 4) = o;
}

// ---------------------------------------------------------------------------
// Kernel 4: out[i] = L2normalize(relu(concat(selfb[i], neigh[i]))).
// One wave per node: 256 elems -> 8 per lane, shuffle-reduce sum of squares.
// ---------------------------------------------------------------------------
__global__ void relu_norm_concat_kernel(const float* __restrict__ selfb,
                                        const float* __restrict__ neigh,
                                        float* __restrict__ out)
{
    const int lane = threadIdx.x & 31;
    const int node = (blockIdx.x * blockDim.x + threadIdx.x) >> 5;
    if (node >= NNODES) return;

    float v[8];
    float ss = 0.f;
#pragma unroll
    for (int t = 0; t < 8; ++t) {
        const int e = lane + 32 * t;                    // 0..255
        float x = (t < 4) ? selfb[(size_t)node * HDIM + e]
                          : neigh[(size_t)node * HDIM + (e - HDIM)];
        x = fmaxf(x, 0.0f);
        v[t] = x;
        ss += x * x;
    }
#pragma unroll
    for (int off = 16; off > 0; off >>= 1)
        ss += __shfl_xor(ss, off, 32);

    const float inv = 1.0f / fmaxf(sqrtf(ss), EPS);
#pragma unroll
    for (int t = 0; t < 8; ++t)
        out[(size_t)node * 256 + lane + 32 * t] = v[t] * inv;
}

// ---------------------------------------------------------------------------
// Launch: build mask once (A read exactly once from HBM), then two GraphSAGE
// layers; all intermediates (<=4 MB each) live in workspace / L2.
// ---------------------------------------------------------------------------
extern "C" void kernel_launch(void* const* d_in, const int* in_sizes, int n_in,
                              void* d_out, int out_size, void* d_ws, size_t ws_size,
                              hipStream_t stream)
{
    const float* x   = (const float*)d_in[0];
    const float* A   = (const float*)d_in[1];
    const float* Wp0 = (const float*)d_in[2];
    const float* bp0 = (const float*)d_in[3];
    const float* Ws0 = (const float*)d_in[4];
    const float* bs0 = (const float*)d_in[5];
    const float* Wn0 = (const float*)d_in[6];
    const float* bn0 = (const float*)d_in[7];
    const float* Wp1 = (const float*)d_in[8];
    const float* bp1 = (const float*)d_in[9];
    const float* Ws1 = (const float*)d_in[10];
    const float* bs1 = (const float*)d_in[11];
    const float* Wn1 = (const float*)d_in[12];
    const float* bn1 = (const float*)d_in[13];

    // Workspace layout (16B-aligned slices)
    const size_t NW    = (size_t)NNODES * (NNODES / 32);   // mask words
    const size_t NH    = (size_t)NNODES * HDIM;            // [N,128] floats
    unsigned* maskw  = (unsigned*)d_ws;
    float*    scores = (float*)((char*)d_ws + NW * 4);
    float*    selfb  = scores + NH;
    float*    pooled = selfb + NH;
    float*    neigh  = pooled + NH;
    float*    x1     = neigh + NH;                         // [N,256]

    // 1) A -> bitmask (the only HBM-heavy pass: 64 MB read, 2 MB written)
    build_mask_kernel<<<1024, 256, 0, stream>>>(A, maskw);

    // GEMM grid: (M/16)*(128/16) waves = 2048 -> 65536 threads
    const dim3 gemmGrid(256), gemmBlk(256);
    const dim3 waveGrid(NNODES / 8), waveBlk(256);         // 4096 waves

    // ---- layer 0 (K = 256) ----
    gemm_bias_act_kernel<<<gemmGrid, gemmBlk, 0, stream>>>(x, Wp0, bp0, scores, NNODES, 256, 1);
    gemm_bias_act_kernel<<<gemmGrid, gemmBlk, 0, stream>>>(x, Ws0, bs0, selfb,  NNODES, 256, 0);
    maxpool_kernel<<<waveGrid, waveBlk, 0, stream>>>(maskw, scores, pooled);
    gemm_bias_act_kernel<<<gemmGrid, gemmBlk, 0, stream>>>(pooled, Wn0, bn0, neigh, NNODES, 128, 0);
    relu_norm_concat_kernel<<<waveGrid, waveBlk, 0, stream>>>(selfb, neigh, x1);

    // ---- layer 1 (K = 256, input x1) ----
    gemm_bias_act_kernel<<<gemmGrid, gemmBlk, 0, stream>>>(x1, Wp1, bp1, scores, NNODES, 256, 1);
    gemm_bias_act_kernel<<<gemmGrid, gemmBlk, 0, stream>>>(x1, Ws1, bs1, selfb,  NNODES, 256, 0);
    maxpool_kernel<<<waveGrid, waveBlk, 0, stream>>>(maskw, scores, pooled);
    gemm_bias_act_kernel<<<gemmGrid, gemmBlk, 0, stream>>>(pooled, Wn1, bn1, neigh, NNODES, 128, 0);
    relu_norm_concat_kernel<<<waveGrid, waveBlk, 0, stream>>>(selfb, neigh, (float*)d_out);
}